// MultiScaleGrouping_50981261803949
// MI455X (gfx1250) — compile-verified
//
#include <hip/hip_runtime.h>
#include <math.h>

typedef __attribute__((ext_vector_type(2))) float v2f;
typedef __attribute__((ext_vector_type(8))) float v8f;

#define B_    8
#define N_    8192
#define MAXC  512

// flat float offsets into d_out (patches0,1,2 then centers0,1,2)
#define OFF_P0 ((size_t)0)
#define OFF_P1 ((size_t)393216)   // + 8*512*32*3
#define OFF_P2 ((size_t)491520)   // + 8*256*16*3
#define OFF_C0 ((size_t)516096)   // + 8*128*8*3
#define OFF_C1 ((size_t)528384)   // + 8*512*3
#define OFF_C2 ((size_t)534528)   // + 8*256*3

// ---------------------------------------------------------------------------
// FPS: one block per batch, 512 threads. The whole point cloud (96 KB) is
// staged into LDS once (coalesced), so the 512 sequential argmax iterations
// (the serial critical path of the pipeline) run entirely out of LDS.
// Emits the reference's scan semantics: out[0]=0, out[t]=argmax after t updates.
// FPS-512 prefixes serve the 256/128 scales too.
// ---------------------------------------------------------------------------
__global__ __launch_bounds__(512)
void fps_kernel(const float* __restrict__ xyz, float* __restrict__ out,
                int* __restrict__ fpsIdx) {
    const int b = blockIdx.x;
    const int t = threadIdx.x;
    const float* px = xyz + (size_t)b * N_ * 3;

    __shared__ float sp[N_ * 3];      // 96 KB interleaved xyz
    __shared__ float sval[512];
    __shared__ int   sidx[512];

    for (int i = t; i < N_ * 3; i += 512) sp[i] = px[i];
    __syncthreads();

    float dmin[16];
#pragma unroll
    for (int i = 0; i < 16; ++i) dmin[i] = 3.4e38f;

    int far = 0;
    for (int it = 0; it < MAXC; ++it) {
        if (t == 0) fpsIdx[b * MAXC + it] = far;
        const float cx = sp[far * 3 + 0];
        const float cy = sp[far * 3 + 1];
        const float cz = sp[far * 3 + 2];
        float bestv = -1.0f;
        int   besti = N_;
#pragma unroll
        for (int i = 0; i < 16; ++i) {
            const int n = t + i * 512;
            const float dx = sp[n * 3 + 0] - cx;
            const float dy = sp[n * 3 + 1] - cy;
            const float dz = sp[n * 3 + 2] - cz;
            const float d  = dx * dx + dy * dy + dz * dz;
            const float m  = fminf(dmin[i], d);
            dmin[i] = m;
            if (m > bestv) { bestv = m; besti = n; }
        }
        sval[t] = bestv; sidx[t] = besti;
        __syncthreads();
        for (int s = 256; s > 0; s >>= 1) {
            if (t < s) {
                const float v = sval[t + s]; const int i2 = sidx[t + s];
                if (v > sval[t] || (v == sval[t] && i2 < sidx[t])) {
                    sval[t] = v; sidx[t] = i2;
                }
            }
            __syncthreads();
        }
        far = sidx[0];
        __syncthreads();  // protect sval/sidx reuse next iteration
    }

    // write all three centers outputs (prefix property)
    const int m = t;
    const int idx = fpsIdx[b * MAXC + m];
    const float x = sp[idx * 3 + 0];
    const float y = sp[idx * 3 + 1];
    const float z = sp[idx * 3 + 2];
    float* c0 = out + OFF_C0 + ((size_t)b * 512 + m) * 3;
    c0[0] = x; c0[1] = y; c0[2] = z;
    if (m < 256) {
        float* c1 = out + OFF_C1 + ((size_t)b * 256 + m) * 3;
        c1[0] = x; c1[1] = y; c1[2] = z;
    }
    if (m < 128) {
        float* c2 = out + OFF_C2 + ((size_t)b * 128 + m) * 3;
        c2[0] = x; c2[1] = y; c2[2] = z;
    }
}

// ---------------------------------------------------------------------------
// kNN + gather + recenter. One block = one batch x 16-center tile.
// Scores S[m][n] = |p_n|^2 - 2*c_m.p_n via V_WMMA_F32_16X16X4_F32:
//   A[m][*] = (-2cx,-2cy,-2cz, 1), B[*][n] = (px,py,pz,|p|^2), C = 0.
// Row-constant |c|^2 dropped (ordering invariant). f32 keeps ref precision.
// Loads for the 4 WMMA tiles per super-chunk are issued as a batch (4
// outstanding global_load_b96) so WMMA/VALU overlap memory latency.
// ---------------------------------------------------------------------------
#define SUPER   512
#define SSTRIDE 513   // 513 % 64 == 1 -> conflict-free column walk per center

__global__ __launch_bounds__(256)
void knn_kernel(const float* __restrict__ xyz, const int* __restrict__ fpsIdx,
                float* __restrict__ out, int M, int K, size_t offP) {
    const int b    = blockIdx.y;
    const int tile = blockIdx.x;
    const int t    = threadIdx.x;
    const int lane = t & 31;
    const int wave = t >> 5;
    const int pos  = lane & 15;
    const int half = lane >> 4;

    const float* px = xyz + (size_t)b * N_ * 3;

    __shared__ float sScore[16 * SSTRIDE];
    __shared__ float sCx[16], sCy[16], sCz[16];
    __shared__ float sKd[16][32];
    __shared__ int   sKi[16][32];

    if (t < 16) {
        const int ci = fpsIdx[b * MAXC + tile * 16 + t];
        sCx[t] = px[ci * 3 + 0];
        sCy[t] = px[ci * 3 + 1];
        sCz[t] = px[ci * 3 + 2];
        for (int j = 0; j < K; ++j) { sKd[t][j] = 3.4e38f; sKi[t][j] = 0; }
    }
    __syncthreads();

    // A fragment: lanes 0-15 hold (K0,K1) of row M=pos, lanes 16-31 hold (K2,K3)
    v2f afrag;
    {
        const float cx = sCx[pos], cy = sCy[pos], cz = sCz[pos];
        afrag.x = half ? (-2.0f * cz) : (-2.0f * cx);
        afrag.y = half ? 1.0f         : (-2.0f * cy);
    }

    for (int sc = 0; sc < N_ / SUPER; ++sc) {
        const int base = sc * SUPER;

        // phase 1: issue all 4 tile loads (kept in flight together)
        float qx[4], qy[4], qz[4];
#pragma unroll
        for (int tt = 0; tt < 4; ++tt) {
            const int nrel = (wave * 4 + tt) * 16 + pos;   // 0..511
            const int n    = base + nrel;
            __builtin_prefetch(px + (size_t)(n + SUPER) * 3, 0, 1);
            qx[tt] = px[n * 3 + 0];
            qy[tt] = px[n * 3 + 1];
            qz[tt] = px[n * 3 + 2];
        }

        // phase 2: WMMA per tile, scores to LDS
#pragma unroll
        for (int tt = 0; tt < 4; ++tt) {
            const int nrel = (wave * 4 + tt) * 16 + pos;
            const float q2 = qx[tt] * qx[tt] + qy[tt] * qy[tt] + qz[tt] * qz[tt];
            // B fragment: lanes 0-15 hold (K0,K1) of col N=pos, lanes 16-31 (K2,K3)
            v2f bfrag;
            bfrag.x = half ? qz[tt] : qx[tt];
            bfrag.y = half ? q2     : qy[tt];
            v8f cacc = {};
            v8f dres = __builtin_amdgcn_wmma_f32_16x16x4_f32(
                false, afrag, false, bfrag, (short)0, cacc, false, false);
#pragma unroll
            for (int r = 0; r < 8; ++r)
                sScore[(r + 8 * half) * SSTRIDE + nrel] = dres[r];
        }
        __syncthreads();

        // per-center serial top-K merge (stable, first-index wins on ties)
        if (t < 16) {
            float worst = sKd[t][K - 1];
            for (int n = 0; n < SUPER; ++n) {
                const float s = sScore[t * SSTRIDE + n];
                if (s < worst) {
                    const int gi = base + n;
                    int j = K - 1;
                    while (j > 0 && sKd[t][j - 1] > s) {
                        sKd[t][j] = sKd[t][j - 1];
                        sKi[t][j] = sKi[t][j - 1];
                        --j;
                    }
                    sKd[t][j] = s;
                    sKi[t][j] = gi;
                    worst = sKd[t][K - 1];
                }
            }
        }
        __syncthreads();
    }

    // gather neighbors, subtract center, store patches
    for (int e = t; e < 16 * K; e += blockDim.x) {
        const int m  = e / K;
        const int j  = e % K;
        const int gi = sKi[m][j];
        const int cg = tile * 16 + m;
        const float ox = px[gi * 3 + 0] - sCx[m];
        const float oy = px[gi * 3 + 1] - sCy[m];
        const float oz = px[gi * 3 + 2] - sCz[m];
        float* o = out + offP + (((size_t)b * M + cg) * K + j) * 3;
        o[0] = ox; o[1] = oy; o[2] = oz;
    }
}

// ---------------------------------------------------------------------------
extern "C" void kernel_launch(void* const* d_in, const int* in_sizes, int n_in,
                              void* d_out, int out_size, void* d_ws, size_t ws_size,
                              hipStream_t stream) {
    const float* xyz = (const float*)d_in[0];
    float* out = (float*)d_out;
    int* fpsIdx = (int*)d_ws;   // 8 * 512 ints

    fps_kernel<<<dim3(B_), dim3(512), 0, stream>>>(xyz, out, fpsIdx);

    knn_kernel<<<dim3(512 / 16, B_), dim3(256), 0, stream>>>(xyz, fpsIdx, out, 512, 32, OFF_P0);
    knn_kernel<<<dim3(256 / 16, B_), dim3(256), 0, stream>>>(xyz, fpsIdx, out, 256, 16, OFF_P1);
    knn_kernel<<<dim3(128 / 16, B_), dim3(256), 0, stream>>>(xyz, fpsIdx, out, 128,  8, OFF_P2);
}